// NNSimilarityChunker_57466662420748
// MI455X (gfx1250) — compile-verified
//
#include <hip/hip_runtime.h>
#include <hip/hip_bf16.h>
#include <math.h>

#define SEQ 256
#define DIM 1024
#define NB  8
#define EPSV 1e-5f

typedef __attribute__((ext_vector_type(2))) float v2f;
typedef __attribute__((ext_vector_type(8))) float v8f;

// ---------------------------------------------------------------------------
// Stage 1: G[b] = X[b] * X[b]^T  via V_WMMA_F32_16X16X4_F32 (full fp32 path).
// One wave computes one 16x16 tile of G. 2048 tiles total (8 batches * 16*16).
//
// f32 16x4 A-matrix layout (ISA 7.12.2): lanes 0-15 hold rows M=0..15 with
// K={0,1} in the two VGPRs; lanes 16-31 hold the same rows with K={2,3}.
// B = X^T, so the B fragment indexes X by the tile's column with the same
// lane pattern. C/D: VGPR v, lanes 0-15 -> (M=v, N=lane); lanes 16-31 ->
// (M=v+8, N=lane-16).
// ---------------------------------------------------------------------------
__global__ __launch_bounds__(256) void gram_wmma_f32(const float* __restrict__ X,
                                                     float* __restrict__ G) {
  const int lane = threadIdx.x & 31;
  const int tile = blockIdx.x * 8 + (threadIdx.x >> 5);   // 0..2047
  const int b  = tile >> 8;
  const int tm = (tile >> 4) & 15;
  const int tn = tile & 15;

  const float* Xb = X + (size_t)b * SEQ * DIM;
  const int half = lane >> 4;          // 0: K 0/1, 1: K 2/3
  const int r    = lane & 15;
  const float* pa = Xb + (size_t)(tm * 16 + r) * DIM + half * 2;
  const float* pb = Xb + (size_t)(tn * 16 + r) * DIM + half * 2;

  v8f acc = {};
#pragma unroll 8
  for (int k0 = 0; k0 < DIM; k0 += 4) {
    v2f a  = *(const v2f*)(pa + k0);
    v2f bb = *(const v2f*)(pb + k0);
    acc = __builtin_amdgcn_wmma_f32_16x16x4_f32(
        /*neg_a=*/false, a, /*neg_b=*/false, bb,
        /*c_mod=*/(short)0, acc, /*reuse_a=*/false, /*reuse_b=*/false);
  }

  float* Gb = G + (size_t)b * SEQ * SEQ;
  const int col = tn * 16 + r;
#pragma unroll
  for (int v = 0; v < 8; ++v) {
    int row = tm * 16 + v + half * 8;
    Gb[row * SEQ + col] = acc[v];
  }
}

// ---------------------------------------------------------------------------
// Stage 2: worst windowed cosine. One wave per (b, i). Lane owns k = lane+32m,
// m=0..7 (full row of 256 in registers). Sweep j upward:
//   num_k += G[j, k]                     (running window dot products)
//   ssq    = sum_{k in [i..j]} num_k     (window squared-sum identity)
//   worst  = min_{k in [i..j]} (num_k/L) / max(sqrt(ssq)/L * ||x_k||, EPS)
// Two wave32 butterfly reductions per j step.
// ---------------------------------------------------------------------------
__global__ __launch_bounds__(256) void worst_sim_sweep(const float* __restrict__ G,
                                                       float* __restrict__ out) {
  const int lane = threadIdx.x & 31;
  const int w = blockIdx.x * 8 + (threadIdx.x >> 5);      // 0..2047
  const int b = w >> 8;
  const int i = w & 255;

  const float* Gb = G + (size_t)b * SEQ * SEQ;
  float* Ob = out + (size_t)b * SEQ * SEQ + (size_t)i * SEQ;

  // invalid prefix (j < i) -> 0, matching reference's `valid` mask
  for (int j = lane; j < i; j += 32) Ob[j] = 0.0f;

  float xn[8], num[8];
#pragma unroll
  for (int m = 0; m < 8; ++m) {
    int k = lane + 32 * m;
    xn[m]  = sqrtf(fmaxf(Gb[(size_t)k * SEQ + k], 0.0f));
    num[m] = 0.0f;
  }

  const bool rowValid = (i >= 1);
  for (int j = i; j < SEQ; ++j) {
    const float* grow = Gb + (size_t)j * SEQ;
    float ssq = 0.0f;
#pragma unroll
    for (int m = 0; m < 8; ++m) {
      int k = lane + 32 * m;
      num[m] += grow[k];                  // coalesced row load
      if (k >= i && k <= j) ssq += num[m];
    }
#pragma unroll
    for (int d = 16; d >= 1; d >>= 1) ssq += __shfl_xor(ssq, d, 32);

    const float L    = (float)(j - i + 1);
    const float invL = 1.0f / L;
    const float mean_norm = sqrtf(fmaxf(ssq, 0.0f)) * invL;

    float cmin = __builtin_inff();
#pragma unroll
    for (int m = 0; m < 8; ++m) {
      int k = lane + 32 * m;
      if (k >= i && k <= j) {
        float c = (num[m] * invL) / fmaxf(mean_norm * xn[m], EPSV);
        cmin = fminf(cmin, c);
      }
    }
#pragma unroll
    for (int d = 16; d >= 1; d >>= 1) cmin = fminf(cmin, __shfl_xor(cmin, d, 32));

    if (lane == 0) Ob[j] = (rowValid && j > i) ? cmin : 0.0f;
  }
}

extern "C" void kernel_launch(void* const* d_in, const int* in_sizes, int n_in,
                              void* d_out, int out_size, void* d_ws, size_t ws_size,
                              hipStream_t stream) {
  const float* X = (const float*)d_in[0];
  float* G   = (float*)d_ws;             // 8*256*256*4 = 2 MB scratch
  float* out = (float*)d_out;            // 8*256*256 floats

  // Stage 1: 2048 tile-waves, 8 waves per 256-thread block.
  gram_wmma_f32<<<256, 256, 0, stream>>>(X, G);
  // Stage 2: 2048 (b,i)-waves, 8 waves per 256-thread block.
  worst_sim_sweep<<<256, 256, 0, stream>>>(G, out);
}